// TinyOnnAttention_2379411882355
// MI455X (gfx1250) — compile-verified
//
#include <hip/hip_runtime.h>
#include <hip/hip_bf16.h>
#include <math.h>

typedef float v2f __attribute__((ext_vector_type(2)));
typedef float v8f __attribute__((ext_vector_type(8)));

#define T_SEQ 2048
#define HID_DIM 1024
#define NH 16
#define DH 64
#define NB 32
#define KMAX_SEL 8

// D = A(16x4) * B(4x16) + C, fp32 WMMA. Layout per CDNA5 ISA 7.12.2:
//  A frag: lane l<16 -> row l, {K0,K1}; lane l>=16 -> row l-16, {K2,K3}
//  B frag: lane l<16 -> col l, {K0,K1}; lane l>=16 -> col l-16, {K2,K3}
//  C/D:    vgpr r, lane l<16 -> (M=r,   N=l); lane l>=16 -> (M=r+8, N=l-16)
__device__ __forceinline__ v8f wmma4(v2f a, v2f b, v8f c) {
  return __builtin_amdgcn_wmma_f32_16x16x4_f32(false, a, false, b, (short)0, c,
                                               false, false);
}

__device__ __forceinline__ float apply_act(float x, int act) {
  if (act == 1) return 0.5f * x * (1.0f + erff(x * 0.70710678118654752f));
  if (act == 2) return 1.0f / (1.0f + __expf(-x));
  return x;
}

// ---------------- register-blocked WMMA f32 GEMM ----------------
// One wave computes a 64(M) x 32(N) macro-tile: 4x2 grid of 16x16 WMMA tiles.
// A frags reused across 2 N-tiles, B frags across 4 M-tiles -> ~1 load/WMMA.
// Requires M%64==0, N%32==0. act: 0 none, 1 gelu, 2 sigmoid.
// headSplit: store C as [N/64][M][64] instead of [M][N].
__global__ void wmma_gemm_f32_blk(const float* __restrict__ A,
                                  const float* __restrict__ B,
                                  const float* __restrict__ bias,
                                  float* __restrict__ C, int M, int N, int K,
                                  int act, int headSplit) {
  int wave = threadIdx.x >> 5;
  int lane = threadIdx.x & 31;
  int wtile = blockIdx.x * (blockDim.x >> 5) + wave;
  int wn = N >> 5;
  int total = (M >> 6) * wn;
  if (wtile >= total) return;
  int tm = wtile / wn, tn = wtile - tm * wn;
  int half = lane >> 4, l = lane & 15;
  int m0 = tm << 6, n0 = tn << 5;

  v8f acc[4][2];
#pragma unroll
  for (int mi = 0; mi < 4; ++mi)
#pragma unroll
    for (int ni = 0; ni < 2; ++ni) acc[mi][ni] = (v8f){};

  const float* ap = A + (size_t)(m0 + l) * K + 2 * half;
  const float* bp = B + (size_t)(2 * half) * N + n0 + l;
  const size_t aRowStep = (size_t)16 * K;  // between M-tiles
  for (int k0 = 0; k0 < K; k0 += 4) {
    v2f a[4], b[2];
#pragma unroll
    for (int mi = 0; mi < 4; ++mi) {
      const float* p = ap + (size_t)mi * aRowStep;
      a[mi].x = p[0];
      a[mi].y = p[1];
    }
#pragma unroll
    for (int ni = 0; ni < 2; ++ni) {
      b[ni].x = bp[ni * 16];
      b[ni].y = bp[(size_t)N + ni * 16];
    }
#pragma unroll
    for (int mi = 0; mi < 4; ++mi)
#pragma unroll
      for (int ni = 0; ni < 2; ++ni)
        acc[mi][ni] = wmma4(a[mi], b[ni], acc[mi][ni]);
    ap += 4;
    bp += (size_t)4 * N;
  }
  float bv[2];
#pragma unroll
  for (int ni = 0; ni < 2; ++ni) bv[ni] = bias ? bias[n0 + ni * 16 + l] : 0.0f;
#pragma unroll
  for (int mi = 0; mi < 4; ++mi)
#pragma unroll
    for (int ni = 0; ni < 2; ++ni) {
      int col = n0 + ni * 16 + l;
#pragma unroll
      for (int r = 0; r < 8; ++r) {
        float x = apply_act(acc[mi][ni][r] + bv[ni], act);
        int m = m0 + mi * 16 + r + 8 * half;
        if (headSplit) {
          int hh = col >> 6, dd = col & 63;
          C[((size_t)hh * M + m) * 64 + dd] = x;
        } else {
          C[(size_t)m * N + col] = x;
        }
      }
    }
}

// ---------------- generic per-tile WMMA f32 GEMM (for N%32 != 0) ----------
__global__ void wmma_gemm_f32(const float* __restrict__ A,
                              const float* __restrict__ B,
                              const float* __restrict__ bias,
                              float* __restrict__ C, int M, int N, int K,
                              int act, int headSplit) {
  int wave = threadIdx.x >> 5;
  int lane = threadIdx.x & 31;
  int tile = blockIdx.x * (blockDim.x >> 5) + wave;
  int tiles_n = N >> 4;
  int total = (M >> 4) * tiles_n;
  if (tile >= total) return;
  int tm = tile / tiles_n, tn = tile - tm * tiles_n;
  int half = lane >> 4, l = lane & 15;
  int arow = (tm << 4) + l;
  int bcol = (tn << 4) + l;
  v8f acc = {};
  const float* ap = A + (size_t)arow * K + 2 * half;
  const float* bp = B + (size_t)(2 * half) * N + bcol;
  for (int k0 = 0; k0 < K; k0 += 4) {
    v2f a, b;
    a.x = ap[0];
    a.y = ap[1];
    b.x = bp[0];
    b.y = bp[N];
    acc = wmma4(a, b, acc);
    ap += 4;
    bp += (size_t)4 * N;
  }
  float bv = bias ? bias[bcol] : 0.0f;
#pragma unroll
  for (int r = 0; r < 8; ++r) {
    float x = apply_act(acc[r] + bv, act);
    int m = (tm << 4) + r + 8 * half;
    if (headSplit) {
      int hh = bcol >> 6, dd = bcol & 63;
      C[((size_t)hh * M + m) * 64 + dd] = x;
    } else {
      C[(size_t)m * N + bcol] = x;
    }
  }
}

// ---------------- sliding-window attention ----------------
// grid = H*32 windows, 128 threads (4 waves). Each wave: 16 query rows.
// scores 64x128 (prev+cur window), causal+lookback mask, softmax, P@V.
__global__ void sliding_window_kernel(const float* __restrict__ q,
                                      const float* __restrict__ k,
                                      const float* __restrict__ v,
                                      float* __restrict__ out) {
  extern __shared__ float lds[];
  float* Ks = lds;              // 128 x 66 (padded, bank-conflict free)
  float* Vs = Ks + 128 * 66;    // 128 x 66
  float* Ps = Vs + 128 * 66;    // 64 x 130
  int h = blockIdx.x >> 5;
  int w = blockIdx.x & 31;
  const float* kh = k + (size_t)h * T_SEQ * DH;
  const float* vh = v + (size_t)h * T_SEQ * DH;
  for (int idx = threadIdx.x; idx < 128 * 32; idx += blockDim.x) {
    int r = idx >> 5, c2 = (idx & 31) << 1;
    int g = (w - 1) * 64 + r;  // look-back window; g<0 -> masked, load zeros
    float2 kk = make_float2(0.f, 0.f), vv = make_float2(0.f, 0.f);
    if (g >= 0) {
      kk = *(const float2*)&kh[(size_t)g * DH + c2];
      vv = *(const float2*)&vh[(size_t)g * DH + c2];
    }
    *(float2*)&Ks[r * 66 + c2] = kk;
    *(float2*)&Vs[r * 66 + c2] = vv;
  }
  __syncthreads();

  int wv = threadIdx.x >> 5, lane = threadIdx.x & 31;
  int half = lane >> 4, l = lane & 15;
  const float* qrow =
      q + ((size_t)h * T_SEQ + w * 64 + wv * 16 + l) * DH + 2 * half;

  v8f s[8] = {};
  for (int k0 = 0; k0 < 64; k0 += 4) {
    v2f a;
    a.x = qrow[k0];
    a.y = qrow[k0 + 1];
#pragma unroll
    for (int nt = 0; nt < 8; ++nt) {
      v2f b;  // B = K^T : B[kk][n] = Ks[n][kk]
      const float* kp = &Ks[(nt * 16 + l) * 66 + k0 + 2 * half];
      b.x = kp[0];
      b.y = kp[1];
      s[nt] = wmma4(a, b, s[nt]);
    }
  }
  float rmax[8], rsum[8];
#pragma unroll
  for (int r = 0; r < 8; ++r) { rmax[r] = -3.0e38f; rsum[r] = 0.f; }
#pragma unroll
  for (int nt = 0; nt < 8; ++nt) {
    int n = nt * 16 + l;
#pragma unroll
    for (int r = 0; r < 8; ++r) {
      int m = wv * 16 + r + 8 * half;  // query row within window
      float x = s[nt][r] * 0.125f;
      bool ok = (m + 64 >= n) && (w > 0 || n >= 64);
      x = ok ? x : -3.0e38f;
      s[nt][r] = x;
      rmax[r] = fmaxf(rmax[r], x);
    }
  }
#pragma unroll
  for (int d = 1; d < 16; d <<= 1)
#pragma unroll
    for (int r = 0; r < 8; ++r) rmax[r] = fmaxf(rmax[r], __shfl_xor(rmax[r], d));
#pragma unroll
  for (int nt = 0; nt < 8; ++nt)
#pragma unroll
    for (int r = 0; r < 8; ++r) {
      float e = __expf(s[nt][r] - rmax[r]);
      s[nt][r] = e;
      rsum[r] += e;
    }
#pragma unroll
  for (int d = 1; d < 16; d <<= 1)
#pragma unroll
    for (int r = 0; r < 8; ++r) rsum[r] += __shfl_xor(rsum[r], d);
  float rinv[8];
#pragma unroll
  for (int r = 0; r < 8; ++r) rinv[r] = 1.0f / rsum[r];
#pragma unroll
  for (int nt = 0; nt < 8; ++nt)
#pragma unroll
    for (int r = 0; r < 8; ++r)
      Ps[(wv * 16 + r + 8 * half) * 130 + nt * 16 + l] = s[nt][r] * rinv[r];
  __syncthreads();

  v8f o[4] = {};
  const float* prow = &Ps[(wv * 16 + l) * 130 + 2 * half];
  for (int k0 = 0; k0 < 128; k0 += 4) {
    v2f a;
    a.x = prow[k0];
    a.y = prow[k0 + 1];
#pragma unroll
    for (int nt = 0; nt < 4; ++nt) {
      v2f b;
      const float* vp = &Vs[(k0 + 2 * half) * 66 + nt * 16 + l];
      b.x = vp[0];
      b.y = vp[66];
      o[nt] = wmma4(a, b, o[nt]);
    }
  }
  float* oh = out + ((size_t)h * T_SEQ + w * 64) * DH;
#pragma unroll
  for (int nt = 0; nt < 4; ++nt)
#pragma unroll
    for (int r = 0; r < 8; ++r)
      oh[(size_t)(wv * 16 + r + 8 * half) * DH + nt * 16 + l] = o[nt][r];
}

// ---------------- compressed branch ----------------
// grid = H * (T/16), 32 threads (1 wave): importance(16x32) = q@kc^T*scale,
// accumulate block-score column sums, softmax over 32, comp = P @ vc.
__global__ void comp_branch_kernel(const float* __restrict__ q,
                                   const float* __restrict__ kc,
                                   const float* __restrict__ vc,
                                   float* __restrict__ bs,
                                   float* __restrict__ comp) {
  __shared__ float Ps[16 * 34];
  int h = blockIdx.x >> 7;
  int tt = blockIdx.x & 127;
  int t0 = tt << 4;
  int lane = threadIdx.x, half = lane >> 4, l = lane & 15;
  const float* qrow = q + ((size_t)h * T_SEQ + t0 + l) * DH + 2 * half;
  v8f s[2] = {};
  for (int k0 = 0; k0 < 64; k0 += 4) {
    v2f a;
    a.x = qrow[k0];
    a.y = qrow[k0 + 1];
#pragma unroll
    for (int nt = 0; nt < 2; ++nt) {
      v2f b;  // B[kk][n] = kc[h*32+n][kk]
      const float* kp = kc + (size_t)(h * 32 + nt * 16 + l) * 64 + k0 + 2 * half;
      b.x = kp[0];
      b.y = kp[1];
      s[nt] = wmma4(a, b, s[nt]);
    }
  }
#pragma unroll
  for (int nt = 0; nt < 2; ++nt)
#pragma unroll
    for (int r = 0; r < 8; ++r) s[nt][r] *= 0.125f;

  int qb = t0 >> 6;  // query block for block-score accumulation
#pragma unroll
  for (int nt = 0; nt < 2; ++nt) {
    float cs = 0.f;
#pragma unroll
    for (int r = 0; r < 8; ++r) cs += s[nt][r];
    cs += __shfl_xor(cs, 16);  // other 8 rows live in the other lane-half
    if (half == 0)
      atomicAdd(&bs[((size_t)h * 32 + qb) * 32 + nt * 16 + l],
                cs * (1.0f / 64.0f));
  }
  float rmax[8], rsum[8];
#pragma unroll
  for (int r = 0; r < 8; ++r) rmax[r] = fmaxf(s[0][r], s[1][r]);
#pragma unroll
  for (int d = 1; d < 16; d <<= 1)
#pragma unroll
    for (int r = 0; r < 8; ++r) rmax[r] = fmaxf(rmax[r], __shfl_xor(rmax[r], d));
#pragma unroll
  for (int r = 0; r < 8; ++r) rsum[r] = 0.f;
#pragma unroll
  for (int nt = 0; nt < 2; ++nt)
#pragma unroll
    for (int r = 0; r < 8; ++r) {
      float e = __expf(s[nt][r] - rmax[r]);
      s[nt][r] = e;
      rsum[r] += e;
    }
#pragma unroll
  for (int d = 1; d < 16; d <<= 1)
#pragma unroll
    for (int r = 0; r < 8; ++r) rsum[r] += __shfl_xor(rsum[r], d);
#pragma unroll
  for (int nt = 0; nt < 2; ++nt)
#pragma unroll
    for (int r = 0; r < 8; ++r)
      Ps[(r + 8 * half) * 34 + nt * 16 + l] = s[nt][r] / rsum[r];
  __syncthreads();

  v8f o[4] = {};
  const float* prow = &Ps[l * 34 + 2 * half];
  for (int k0 = 0; k0 < 32; k0 += 4) {
    v2f a;
    a.x = prow[k0];
    a.y = prow[k0 + 1];
#pragma unroll
    for (int nt = 0; nt < 4; ++nt) {
      v2f b;
      const float* vp = vc + (size_t)(h * 32 + k0 + 2 * half) * 64 + nt * 16 + l;
      b.x = vp[0];
      b.y = vp[64];
      o[nt] = wmma4(a, b, o[nt]);
    }
  }
  float* oh = comp + ((size_t)h * T_SEQ + t0) * DH;
#pragma unroll
  for (int nt = 0; nt < 4; ++nt)
#pragma unroll
    for (int r = 0; r < 8; ++r)
      oh[(size_t)(r + 8 * half) * DH + nt * 16 + l] = o[nt][r];
}

// ---------------- block stats: softmax/entropy/top-k/mask ----------------
__global__ void block_stats_kernel(const float* __restrict__ bs,
                                   const float* __restrict__ ksc,
                                   const float* __restrict__ kbi,
                                   unsigned char* __restrict__ mask,
                                   float* __restrict__ avg_acc,
                                   float* __restrict__ tail) {
  int h = blockIdx.x, j = threadIdx.x;  // one row per thread
  const float* row = bs + ((size_t)h * 32 + j) * 32;
  float sc[32], p[32];
  float mx = -3.0e38f;
  for (int c = 0; c < 32; ++c) { sc[c] = row[c]; mx = fmaxf(mx, sc[c]); }
  float ssum = 0.f;
  for (int c = 0; c < 32; ++c) { p[c] = __expf(sc[c] - mx); ssum += p[c]; }
  float inv = 1.0f / ssum;
  float ent = 0.f;
  for (int c = 0; c < 32; ++c) {
    float pp = p[c] * inv;
    ent -= pp * __logf(pp + 1e-9f);
  }
  float ne = ent / (logf(32.0f) + 1e-9f);
  float kr = 1.0f / (1.0f + __expf(-(ksc[h] * ne + kbi[h])));
  float dkf = fminf(fmaxf(8.0f * kr, 1.0f), 8.0f);
  int dk = (int)dkf;
  tail[1 + h * 32 + j] = kr;         // k_ratio
  tail[1 + 512 + h * 32 + j] = ne;   // norm_entropy
  atomicAdd(avg_acc, (float)dk);
  unsigned used = 0;
  unsigned char mrow[32];
  for (int c = 0; c < 32; ++c) mrow[c] = 0;
  mrow[j] = 1;  // eye
  for (int kk = 0; kk < KMAX_SEL; ++kk) {
    float best = -3.4e38f;
    int bi = 0;
    for (int c = 0; c < 32; ++c)
      if (!((used >> c) & 1u) && sc[c] > best) { best = sc[c]; bi = c; }
    used |= 1u << bi;
    if (kk < dk) mrow[bi] = 1;
  }
  for (int c = 0; c < 32; ++c) mask[((size_t)h * 32 + j) * 32 + c] = mrow[c];
}

// ---------------- fine block-sparse attention (flash over selected blocks) --
__global__ void fine_attn_kernel(const float* __restrict__ q,
                                 const float* __restrict__ k,
                                 const float* __restrict__ v,
                                 const unsigned char* __restrict__ mask,
                                 float* __restrict__ out) {
  __shared__ float Ks[64 * 66];
  __shared__ float Vs[64 * 66];
  __shared__ float Ps[64 * 66];
  int h = blockIdx.x >> 5, qb = blockIdx.x & 31;
  int wv = threadIdx.x >> 5, lane = threadIdx.x & 31;
  int half = lane >> 4, l = lane & 15;
  const float* kh = k + (size_t)h * T_SEQ * DH;
  const float* vh = v + (size_t)h * T_SEQ * DH;
  const float* qrow =
      q + ((size_t)h * T_SEQ + qb * 64 + wv * 16 + l) * DH + 2 * half;
  const unsigned char* mrow = mask + ((size_t)h * 32 + qb) * 32;
  v8f o[4] = {};
  float rmax[8], rsum[8];
#pragma unroll
  for (int r = 0; r < 8; ++r) { rmax[r] = -3.0e38f; rsum[r] = 0.f; }

  for (int j = 0; j < 32; ++j) {
    if (!mrow[j]) continue;  // uniform across block
    __syncthreads();
    for (int idx = threadIdx.x; idx < 64 * 32; idx += blockDim.x) {
      int r = idx >> 5, c2 = (idx & 31) << 1;
      *(float2*)&Ks[r * 66 + c2] =
          *(const float2*)&kh[(size_t)(j * 64 + r) * DH + c2];
      *(float2*)&Vs[r * 66 + c2] =
          *(const float2*)&vh[(size_t)(j * 64 + r) * DH + c2];
    }
    __syncthreads();
    v8f s[4] = {};
    for (int k0 = 0; k0 < 64; k0 += 4) {
      v2f a;
      a.x = qrow[k0];
      a.y = qrow[k0 + 1];
#pragma unroll
      for (int nt = 0; nt < 4; ++nt) {
        v2f b;
        const float* kp = &Ks[(nt * 16 + l) * 66 + k0 + 2 * half];
        b.x = kp[0];
        b.y = kp[1];
        s[nt] = wmma4(a, b, s[nt]);
      }
    }
    float nm[8], ps[8], fac[8];
#pragma unroll
    for (int r = 0; r < 8; ++r) nm[r] = rmax[r];
#pragma unroll
    for (int nt = 0; nt < 4; ++nt)
#pragma unroll
      for (int r = 0; r < 8; ++r) {
        float x = s[nt][r] * 0.125f;
        s[nt][r] = x;
        nm[r] = fmaxf(nm[r], x);
      }
#pragma unroll
    for (int d = 1; d < 16; d <<= 1)
#pragma unroll
      for (int r = 0; r < 8; ++r) nm[r] = fmaxf(nm[r], __shfl_xor(nm[r], d));
#pragma unroll
    for (int r = 0; r < 8; ++r) { fac[r] = __expf(rmax[r] - nm[r]); ps[r] = 0.f; }
#pragma unroll
    for (int nt = 0; nt < 4; ++nt)
#pragma unroll
      for (int r = 0; r < 8; ++r) {
        float e = __expf(s[nt][r] - nm[r]);
        s[nt][r] = e;
        ps[r] += e;
      }
#pragma unroll
    for (int d = 1; d < 16; d <<= 1)
#pragma unroll
      for (int r = 0; r < 8; ++r) ps[r] += __shfl_xor(ps[r], d);
#pragma unroll
    for (int r = 0; r < 8; ++r) {
      rsum[r] = rsum[r] * fac[r] + ps[r];
      rmax[r] = nm[r];
    }
#pragma unroll
    for (int nt = 0; nt < 4; ++nt)
#pragma unroll
      for (int r = 0; r < 8; ++r)
        Ps[(wv * 16 + r + 8 * half) * 66 + nt * 16 + l] = s[nt][r];
#pragma unroll
    for (int nt = 0; nt < 4; ++nt)
#pragma unroll
      for (int r = 0; r < 8; ++r) o[nt][r] *= fac[r];
    __syncthreads();
    const float* prow = &Ps[(wv * 16 + l) * 66 + 2 * half];
    for (int k0 = 0; k0 < 64; k0 += 4) {
      v2f a;
      a.x = prow[k0];
      a.y = prow[k0 + 1];
#pragma unroll
      for (int nt = 0; nt < 4; ++nt) {
        v2f b;
        const float* vp = &Vs[(k0 + 2 * half) * 66 + nt * 16 + l];
        b.x = vp[0];
        b.y = vp[66];
        o[nt] = wmma4(a, b, o[nt]);
      }
    }
  }
  float* oh = out + ((size_t)h * T_SEQ + qb * 64) * DH;
#pragma unroll
  for (int nt = 0; nt < 4; ++nt)
#pragma unroll
    for (int r = 0; r < 8; ++r)
      oh[(size_t)(wv * 16 + r + 8 * half) * DH + nt * 16 + l] =
          o[nt][r] / rsum[r];
}

// ---------------- gate combine ----------------
__global__ void combine_kernel(const float* __restrict__ sw,
                               const float* __restrict__ comp,
                               const float* __restrict__ fine,
                               const float* __restrict__ gate,
                               float* __restrict__ merged) {
  int idx = blockIdx.x * blockDim.x + threadIdx.x;
  if (idx >= T_SEQ * HID_DIM) return;
  int t = idx >> 10, c = idx & 1023, hh = c >> 6, dd = c & 63;
  size_t a = ((size_t)hh * T_SEQ + t) * DH + dd;
  const float* g = gate + (size_t)t * 48 + hh * 3;
  merged[idx] = sw[a] * g[0] + comp[a] * g[1] + fine[a] * g[2];
}

__global__ void zero_kernel(float* __restrict__ p, int n) {
  int i = blockIdx.x * blockDim.x + threadIdx.x;
  if (i < n) p[i] = 0.f;
}

__global__ void finalize_kernel(const float* __restrict__ avg_acc,
                                float* __restrict__ tail) {
  if (threadIdx.x == 0) tail[0] = avg_acc[0] * (1.0f / 512.0f);
}

extern "C" void kernel_launch(void* const* d_in, const int* in_sizes, int n_in,
                              void* d_out, int out_size, void* d_ws,
                              size_t ws_size, hipStream_t stream) {
  const float* X = (const float*)d_in[0];
  const float* Wq = (const float*)d_in[1];
  const float* Wk = (const float*)d_in[2];
  const float* Wv = (const float*)d_in[3];
  const float* Wo = (const float*)d_in[4];
  const float* Wc = (const float*)d_in[5];
  const float* bc = (const float*)d_in[6];
  const float* ksc = (const float*)d_in[7];
  const float* kbi = (const float*)d_in[8];
  const float* Wg = (const float*)d_in[9];
  const float* bg = (const float*)d_in[10];
  float* out = (float*)d_out;

  const size_t U = (size_t)NH * T_SEQ * DH;  // 2,097,152 floats per [H][T][D]
  float* ws = (float*)d_ws;
  float* q = ws;
  float* kbuf = ws + U;
  float* vbuf = ws + 2 * U;
  float* sw = ws + 3 * U;
  float* comp = ws + 4 * U;
  float* fine = ws + 5 * U;
  float* merged = q;  // q is dead after fine attention; reuse
  float* extra = ws + 6 * U;
  float* gate = extra;                    // 2048*48
  float* kc = gate + (size_t)T_SEQ * 48;  // 512*64
  float* vc = kc + 512 * 64;              // 512*64
  float* bs = vc + 512 * 64;              // 16*32*32
  float* avg = bs + 16384;                // 1 (zeroed together with bs)
  unsigned char* mask = (unsigned char*)(avg + 64);  // 16KB

  zero_kernel<<<(16385 + 255) / 256, 256, 0, stream>>>(bs, 16385);

  {  // Q/K/V projections into head-major [H][T][64]; 64x32 macro-tiles
    int wtiles = (T_SEQ / 64) * (HID_DIM / 32);  // 1024 waves
    int blocks = (wtiles + 7) / 8;
    wmma_gemm_f32_blk<<<blocks, 256, 0, stream>>>(X, Wq, nullptr, q, T_SEQ,
                                                  HID_DIM, HID_DIM, 0, 1);
    wmma_gemm_f32_blk<<<blocks, 256, 0, stream>>>(X, Wk, nullptr, kbuf, T_SEQ,
                                                  HID_DIM, HID_DIM, 0, 1);
    wmma_gemm_f32_blk<<<blocks, 256, 0, stream>>>(X, Wv, nullptr, vbuf, T_SEQ,
                                                  HID_DIM, HID_DIM, 0, 1);
  }
  {  // gate = sigmoid(X @ Wg + bg), [2048][48] (N=48 -> per-tile kernel)
    int tiles = (T_SEQ / 16) * (48 / 16);
    wmma_gemm_f32<<<(tiles + 7) / 8, 256, 0, stream>>>(X, Wg, bg, gate, T_SEQ,
                                                       48, HID_DIM, 2, 0);
  }
  sliding_window_kernel<<<NH * NB, 128,
                          (2 * 128 * 66 + 64 * 130) * sizeof(float), stream>>>(
      q, kbuf, vbuf, sw);
  {  // compress MLP: [512][4096] @ Wc + bc, gelu (k rows are contiguous blocks)
    int wtiles = (512 / 64) * (64 / 32);  // 16 waves
    wmma_gemm_f32_blk<<<(wtiles + 7) / 8, 256, 0, stream>>>(kbuf, Wc, bc, kc,
                                                            512, 64, 4096, 1, 0);
    wmma_gemm_f32_blk<<<(wtiles + 7) / 8, 256, 0, stream>>>(vbuf, Wc, bc, vc,
                                                            512, 64, 4096, 1, 0);
  }
  comp_branch_kernel<<<NH * (T_SEQ / 16), 32, 0, stream>>>(q, kc, vc, bs, comp);
  block_stats_kernel<<<NH, 32, 0, stream>>>(bs, ksc, kbi, mask, avg,
                                            out + (size_t)T_SEQ * HID_DIM);
  fine_attn_kernel<<<NH * NB, 128, 0, stream>>>(q, kbuf, vbuf, mask, fine);
  combine_kernel<<<(T_SEQ * HID_DIM + 255) / 256, 256, 0, stream>>>(
      sw, comp, fine, gate, merged);
  {  // output projection
    int wtiles = (T_SEQ / 64) * (HID_DIM / 32);
    wmma_gemm_f32_blk<<<(wtiles + 7) / 8, 256, 0, stream>>>(
        merged, Wo, nullptr, out, T_SEQ, HID_DIM, HID_DIM, 0, 0);
  }
  finalize_kernel<<<1, 1, 0, stream>>>(avg, out + (size_t)T_SEQ * HID_DIM);
  (void)in_sizes;
  (void)n_in;
  (void)out_size;
  (void)ws_size;
}